// StochasticOutputNeuronCell_24592982737427
// MI455X (gfx1250) — compile-verified
//
#include <hip/hip_runtime.h>
#include <cstdint>
#include <cstddef>

#define NEURONS 2048
#define THREADS 256
#define NWAVES  (THREADS / 32)

// reference constants
#define K_INHIB_INC   5.0f
#define K_DECAY       0.9f            // 1 - DECAY_RATE*DT = 1 - 100*0.001
#define K_SIGMA_SQDT  0.15811388f     // DECAY_SIGMA * sqrt(DT)
#define K_DT          0.001f
#define K_LOGCLIP     46.0517018599f  // log(1e20)

typedef unsigned int u32x4 __attribute__((ext_vector_type(4)));
typedef int          i32x8 __attribute__((ext_vector_type(8)));
typedef int          i32x4 __attribute__((ext_vector_type(4)));
typedef float        f32x4 __attribute__((ext_vector_type(4)));

// ---- per-row RNG (runs 3x per row; cost negligible) ----
__device__ __forceinline__ uint64_t splitmix64(uint64_t x) {
  x += 0x9E3779B97F4A7C15ull;
  x = (x ^ (x >> 30)) * 0xBF58476D1CE4E5B9ull;
  x = (x ^ (x >> 27)) * 0x94D049BB133111EBull;
  return x ^ (x >> 31);
}
__device__ __forceinline__ float u01_64(uint64_t key) {
  uint32_t h = (uint32_t)(splitmix64(key) >> 40);
  return ((float)h + 0.5f) * (1.0f / 16777216.0f);
}

// ---- per-element RNG: 5-op 32-bit avalanche (lowbias32) — keeps kernel
//      under the HBM roofline instead of 64-bit-multiply VALU-bound ----
__device__ __forceinline__ float u01_32(uint32_t h) {
  h ^= h >> 16;  h *= 0x21F0AAADu;
  h ^= h >> 15;  h *= 0xD35A2D97u;
  h ^= h >> 15;
  // top 24 bits -> (0,1)
  return ((float)(h >> 8) + 0.5f) * (1.0f / 16777216.0f);
}

__global__ void __launch_bounds__(THREADS)
snn_step_kernel(const float* __restrict__ inputs,
                const float* __restrict__ inhibition,
                float* __restrict__ out_spikes,
                float* __restrict__ out_inhib) {
  __shared__ float lds_row[NEURONS];   // 8 KiB row staging (TDM destination)
  __shared__ float s_sum[NWAVES];
  __shared__ float s_max[NWAVES];
  __shared__ int   s_idx[NWAVES];
  __shared__ float s_spike;
  __shared__ int   s_argmax;

  const int row  = blockIdx.x;
  const int tid  = threadIdx.x;
  const int wave = tid >> 5;
  const int lane = tid & 31;

  // ---- Stage one 2048-float row Global -> LDS via the Tensor Data Mover ----
  if (wave == 0) {
    uint64_t gaddr  = (uint64_t)(uintptr_t)(inputs + (size_t)row * NEURONS);
    uint32_t ldsoff = (uint32_t)(uintptr_t)(&lds_row[0]);

    u32x4 g0;
    g0.x = 1u;                                   // count=1 (valid user descriptor)
    g0.y = ldsoff;                               // lds_addr (bytes)
    g0.z = (uint32_t)(gaddr & 0xFFFFFFFFu);      // global_addr[31:0]
    g0.w = (uint32_t)((gaddr >> 32) & 0x1FFFFFFu) | (2u << 30); // [56:32] | type=2

    i32x8 g1;
    g1[0] = 0x00020000;                 // wg_mask=0, data_size=2 (4B), no pad/iter
    g1[1] = (int)(NEURONS << 16);       // barrier_addr=0 | tensor_dim0[15:0]=2048
    g1[2] = (int)(1 << 16);             // tensor_dim0[31:16]=0 | tensor_dim1[15:0]=1
    g1[3] = (int)(NEURONS << 16);       // tensor_dim1[31:16]=0 | tile_dim0=2048
    g1[4] = 1;                          // tile_dim1=1 | tile_dim2=0
    g1[5] = NEURONS;                    // tensor_dim0_stride[31:0]=2048
    g1[6] = (int)(NEURONS << 16);       // stride0[47:32]=0 | tensor_dim1_stride[15:0]
    g1[7] = 0;

    i32x4 gz = {0, 0, 0, 0};
#if defined(__clang_major__) && (__clang_major__ >= 23)
    i32x8 gz8 = {0, 0, 0, 0, 0, 0, 0, 0};
    __builtin_amdgcn_tensor_load_to_lds(g0, g1, gz, gz, gz8, 0);
#else
    __builtin_amdgcn_tensor_load_to_lds(g0, g1, gz, gz, 0);
#endif
    __builtin_amdgcn_s_wait_tensorcnt(0);
  }
  __syncthreads();  // TDM data visible to all 8 waves

  // ---- Inhibition decay + diffusion noise (per-row; redundant per thread) ----
  const float u1 = u01_64(((uint64_t)0xA11CE50ull << 32) ^ (uint64_t)(2 * row));
  const float u2 = u01_64(((uint64_t)0xA11CE50ull << 32) ^ (uint64_t)(2 * row + 1));
  const float gauss = __fsqrt_rn(-2.0f * __logf(u1)) * __cosf(6.28318530718f * u2);
  const float inhib_new = K_DECAY * inhibition[row] + K_SIGMA_SQDT * gauss;

  // ---- Per-element: clipped-rate sum + Gumbel-max key (categorical) ----
  float local_sum = 0.0f;
  float best      = -__builtin_inff();
  int   besti     = 0;

  const uint32_t rowbase = (uint32_t)row * (uint32_t)NEURONS;
  const f32x4* rowv = (const f32x4*)lds_row;
#pragma unroll
  for (int c = 0; c < 2; ++c) {
    const int v4   = tid + c * THREADS;     // 512 float4 per row
    const int base = v4 * 4;
    const f32x4 x4 = rowv[v4];
    const float xs[4] = {x4.x, x4.y, x4.z, x4.w};
#pragma unroll
    for (int k = 0; k < 4; ++k) {
      const int col = base + k;
      const float z  = xs[k] - inhib_new;
      const float zc = fminf(fmaxf(z, -K_LOGCLIP), K_LOGCLIP); // log(clip(exp z))
      local_sum += __expf(zc);                                 // clip(exp z,1e-20,1e20)
      const float ug  = u01_32((rowbase + (uint32_t)col) ^ 0x9E3779B9u);
      const float key = zc - __logf(-__logf(ug));              // logit + Gumbel
      if (key > best) { best = key; besti = col; }
    }
  }

  // ---- wave32 reduction: sum and (max, first-index) ----
#pragma unroll
  for (int off = 16; off > 0; off >>= 1) {
    local_sum += __shfl_xor(local_sum, off, 32);
    const float ov = __shfl_xor(best,  off, 32);
    const int   oi = __shfl_xor(besti, off, 32);
    if (ov > best || (ov == best && oi < besti)) { best = ov; besti = oi; }
  }
  if (lane == 0) { s_sum[wave] = local_sum; s_max[wave] = best; s_idx[wave] = besti; }
  __syncthreads();

  if (tid == 0) {
    float tot = 0.0f;
    float b   = s_max[0];
    int   bi  = s_idx[0];
#pragma unroll
    for (int w = 0; w < NWAVES; ++w) {
      tot += s_sum[w];
      if (s_max[w] > b || (s_max[w] == b && s_idx[w] < bi)) { b = s_max[w]; bi = s_idx[w]; }
    }
    const float ur    = u01_64(((uint64_t)0x5B1CEull << 40) ^ (uint64_t)row);
    const float spike = (ur < K_DT * tot) ? 1.0f : 0.0f;
    s_spike  = spike;
    s_argmax = bi;
    out_inhib[row] = inhib_new + spike * K_INHIB_INC;   // inhib + max(out)*5
  }
  __syncthreads();

  // ---- write one-hot * spike; pure streaming -> non-temporal 128b stores ----
  const float spike = s_spike;
  const int   arg   = s_argmax;
  f32x4* outv = (f32x4*)(out_spikes + (size_t)row * NEURONS);
#pragma unroll
  for (int c = 0; c < 2; ++c) {
    const int v4   = tid + c * THREADS;
    const int base = v4 * 4;
    f32x4 o;
    o.x = (base + 0 == arg) ? spike : 0.0f;
    o.y = (base + 1 == arg) ? spike : 0.0f;
    o.z = (base + 2 == arg) ? spike : 0.0f;
    o.w = (base + 3 == arg) ? spike : 0.0f;
    __builtin_nontemporal_store(o, outv + v4);
  }
}

extern "C" void kernel_launch(void* const* d_in, const int* in_sizes, int n_in,
                              void* d_out, int out_size, void* d_ws, size_t ws_size,
                              hipStream_t stream) {
  const float* inputs     = (const float*)d_in[0];  // [B, N] f32
  const float* inhibition = (const float*)d_in[1];  // [B, 1] f32
  const int batch = in_sizes[1];                    // 8192 rows
  float* out_spikes = (float*)d_out;                           // [B, N]
  float* out_inhib  = (float*)d_out + (size_t)batch * NEURONS; // [B, 1]

  snn_step_kernel<<<batch, THREADS, 0, stream>>>(inputs, inhibition,
                                                 out_spikes, out_inhib);
}